// QErrorMitigationModel_13700945674524
// MI455X (gfx1250) — compile-verified
//
#include <hip/hip_runtime.h>
#include <hip/hip_bf16.h>

// ---------------------------------------------------------------------------
// GATv2 GNN forward for MI455X (gfx1250, wave32).
//  - Node transforms (x@W+b) use V_WMMA_F32_16X16X4_F32 (exact fp32 matmul),
//    templated on K/NOUT so the inner loop has no exec-mask guards.
//  - Edge gather/softmax/scatter phase is L2-resident (xl/xr ~25.6MB << 192MB
//    L2), one wave32 per edge, float4 (b128) gathers + f32 atomics.
// ---------------------------------------------------------------------------

typedef __attribute__((ext_vector_type(2))) float v2f;
typedef __attribute__((ext_vector_type(8))) float v8f;

#define N_NODES 50000
#define N_EDGES 800000
#define N_GRAPHS 64

// ---------------- float atomic max via int/uint ordering trick -------------
__device__ __forceinline__ void atomicMaxFloat(float* addr, float value) {
    if (value >= 0.0f) {
        atomicMax((int*)addr, __float_as_int(value));
    } else {
        atomicMin((unsigned int*)addr, __float_as_uint(value));
    }
}

__device__ __forceinline__ v8f wmma_f32(v2f a, v2f b, v8f c) {
    return __builtin_amdgcn_wmma_f32_16x16x4_f32(
        /*neg_a=*/false, a, /*neg_b=*/false, b,
        /*c_mod=*/(short)0, c, /*reuse_a=*/false, /*reuse_b=*/false);
}

// ---------------- WMMA GEMM: out{l,r} = X @ W{l,r} + b{l,r} ----------------
// One wave (32 threads) per 16x16 output tile; the wave computes BOTH the L
// and R transforms, sharing the A-operand loads.
// grid = (Mrows/16, NOUT/16), block = 32.
// A (16x4 f32) lane layout: lanes 0-15 M=0..15 {K=k0,k0+1}; lanes 16-31 {k0+2,k0+3}
// B (4x16 f32): row-striped across lanes (mirrors A). C/D: 8 VGPRs per ISA table.
template<int K, int NOUT>
__global__ void gemm_wmma_lr(const float* __restrict__ X,
                             const float* __restrict__ Wl, const float* __restrict__ bl,
                             const float* __restrict__ Wr, const float* __restrict__ br,
                             float* __restrict__ outl, float* __restrict__ outr)
{
    const int lane  = threadIdx.x;           // 0..31
    const int m0    = blockIdx.x * 16;
    const int n0    = blockIdx.y * 16;
    const int mrow  = m0 + (lane & 15);
    const int khalf = (lane >> 4) * 2;       // 0 for lanes 0-15, 2 for 16-31
    const int ncol  = n0 + (lane & 15);
    const float* __restrict__ Xrow = X + (size_t)mrow * K;

    v8f accL = {}, accR = {};

    constexpr int FULL = K / 4;              // unguarded K-steps
#pragma unroll
    for (int s = 0; s < FULL; ++s) {
        const int ka = s * 4 + khalf;
        v2f a, bL, bR;
        a.x  = Xrow[ka];
        a.y  = Xrow[ka + 1];
        bL.x = Wl[(size_t)(ka)     * NOUT + ncol];
        bL.y = Wl[(size_t)(ka + 1) * NOUT + ncol];
        bR.x = Wr[(size_t)(ka)     * NOUT + ncol];
        bR.y = Wr[(size_t)(ka + 1) * NOUT + ncol];
        accL = wmma_f32(a, bL, accL);
        accR = wmma_f32(a, bR, accR);
    }
    if constexpr ((K % 4) != 0) {            // single zero-padded tail step
        const int ka = FULL * 4 + khalf;
        v2f a = {}, bL = {}, bR = {};
        if (ka < K) {
            a.x  = Xrow[ka];
            bL.x = Wl[(size_t)ka * NOUT + ncol];
            bR.x = Wr[(size_t)ka * NOUT + ncol];
        }
        if (ka + 1 < K) {
            a.y  = Xrow[ka + 1];
            bL.y = Wl[(size_t)(ka + 1) * NOUT + ncol];
            bR.y = Wr[(size_t)(ka + 1) * NOUT + ncol];
        }
        accL = wmma_f32(a, bL, accL);
        accR = wmma_f32(a, bR, accR);
    }

    // C/D layout: VGPR r -> M = r (+8 for lanes 16-31), N = lane&15
    const float bvL   = bl[ncol];
    const float bvR   = br[ncol];
    const int   rbase = m0 + ((lane >> 4) ? 8 : 0);
#pragma unroll
    for (int r = 0; r < 8; ++r) {
        outl[(size_t)(rbase + r) * NOUT + ncol] = accL[r] + bvL;
        outr[(size_t)(rbase + r) * NOUT + ncol] = accR[r] + bvR;
    }
}

// ---------------- per-layer init: nmax=-BIG, nden=0, hout=0 ----------------
__global__ void init_layer(float* __restrict__ nmax, float* __restrict__ nden,
                           float* __restrict__ hout, int nNodes, int D)
{
    const int i = blockIdx.x * blockDim.x + threadIdx.x;
    if (i < nNodes * D) hout[i] = 0.0f;
    if (i < nNodes) { nmax[i] = -3.402823466e38f; nden[i] = 0.0f; }
}

// ---------------- edge scores + segment max --------------------------------
// score[e] = att . leaky_relu(xl[src] + xr[dst], 0.2); wave32 per edge.
// D==128: each lane handles 4 contiguous dims via one b128 load per operand.
// D==32 : each lane handles 1 dim.
template<int D>
__global__ void edge_score(const float* __restrict__ xl, const float* __restrict__ xr,
                           const float* __restrict__ att, const int* __restrict__ ei,
                           float* __restrict__ esc, float* __restrict__ nmax,
                           int nEdges, int nNodes)
{
    const int e    = blockIdx.x * (blockDim.x >> 5) + (threadIdx.x >> 5);
    const int lane = threadIdx.x & 31;
    if (e >= nEdges + nNodes) return;
    if (e + 4096 < nEdges) {                          // stream-ahead prefetch
        __builtin_prefetch(&ei[e + 4096], 0, 1);
        __builtin_prefetch(&ei[nEdges + e + 4096], 0, 1);
    }
    int s, d;
    if (e < nEdges) { s = ei[e]; d = ei[nEdges + e]; }
    else            { s = d = e - nEdges; }            // self-loop

    float sum = 0.0f;
    if constexpr (D == 128) {
        const float4 a  = ((const float4*)(xl + (size_t)s * D))[lane];
        const float4 b  = ((const float4*)(xr + (size_t)d * D))[lane];
        const float4 at = ((const float4*)att)[lane];
        float v0 = a.x + b.x, v1 = a.y + b.y, v2 = a.z + b.z, v3 = a.w + b.w;
        v0 = (v0 > 0.0f) ? v0 : 0.2f * v0;
        v1 = (v1 > 0.0f) ? v1 : 0.2f * v1;
        v2 = (v2 > 0.0f) ? v2 : 0.2f * v2;
        v3 = (v3 > 0.0f) ? v3 : 0.2f * v3;
        sum = v0 * at.x + v1 * at.y + v2 * at.z + v3 * at.w;
    } else {
        float v = xl[(size_t)s * D + lane] + xr[(size_t)d * D + lane];
        v = (v > 0.0f) ? v : 0.2f * v;
        sum = v * att[lane];
    }
#pragma unroll
    for (int off = 16; off > 0; off >>= 1) sum += __shfl_down(sum, off, 32);
    if (lane == 0) {
        esc[e] = sum;
        atomicMaxFloat(&nmax[d], sum);
    }
}

// ---------------- exp(score - max) + segment-sum denom ---------------------
__global__ void edge_exp(const int* __restrict__ ei, const float* __restrict__ nmax,
                         float* __restrict__ esc, float* __restrict__ nden,
                         int nEdges, int nNodes)
{
    const int e = blockIdx.x * blockDim.x + threadIdx.x;
    if (e >= nEdges + nNodes) return;
    const int d = (e < nEdges) ? ei[nEdges + e] : (e - nEdges);
    const float ex = expf(esc[e] - nmax[d]);
    esc[e] = ex;
    atomicAdd(&nden[d], ex);
}

// ---------------- out[dst] += alpha * xl[src] ------------------------------
template<int D>
__global__ void edge_aggregate(const float* __restrict__ xl, const int* __restrict__ ei,
                               const float* __restrict__ esc, const float* __restrict__ nden,
                               float* __restrict__ out, int nEdges, int nNodes)
{
    const int e    = blockIdx.x * (blockDim.x >> 5) + (threadIdx.x >> 5);
    const int lane = threadIdx.x & 31;
    if (e >= nEdges + nNodes) return;
    if (e + 4096 < nEdges) {
        __builtin_prefetch(&ei[e + 4096], 0, 1);
        __builtin_prefetch(&ei[nEdges + e + 4096], 0, 1);
    }
    int s, d;
    if (e < nEdges) { s = ei[e]; d = ei[nEdges + e]; }
    else            { s = d = e - nEdges; }
    const float alpha = esc[e] / (nden[d] + 1e-16f);
    if constexpr (D == 128) {
        const float4 a = ((const float4*)(xl + (size_t)s * D))[lane];
        float* o = out + (size_t)d * D + lane * 4;
        atomicAdd(o + 0, alpha * a.x);
        atomicAdd(o + 1, alpha * a.y);
        atomicAdd(o + 2, alpha * a.z);
        atomicAdd(o + 3, alpha * a.w);
    } else {
        atomicAdd(&out[(size_t)d * D + lane], alpha * xl[(size_t)s * D + lane]);
    }
}

// ---------------- h = elu(h + bias), float4 elementwise --------------------
__global__ void bias_elu4(float4* __restrict__ h, const float* __restrict__ bias,
                          int n4, int D)
{
    const int i = blockIdx.x * blockDim.x + threadIdx.x;
    if (i >= n4) return;
    const int c = (i * 4) % D;                 // D % 4 == 0, chunk within a row
    float4 v = h[i];
    v.x += bias[c];     v.y += bias[c + 1];
    v.z += bias[c + 2]; v.w += bias[c + 3];
    v.x = (v.x > 0.0f) ? v.x : (expf(v.x) - 1.0f);
    v.y = (v.y > 0.0f) ? v.y : (expf(v.y) - 1.0f);
    v.z = (v.z > 0.0f) ? v.z : (expf(v.z) - 1.0f);
    v.w = (v.w > 0.0f) ? v.w : (expf(v.w) - 1.0f);
    h[i] = v;
}

// ---------------- pool init + segment mean pool ----------------------------
__global__ void init_pool(float* __restrict__ gsum, float* __restrict__ gcnt, int G)
{
    const int i = blockIdx.x * blockDim.x + threadIdx.x;
    if (i < G * 32) gsum[i] = 0.0f;
    if (i < G)      gcnt[i] = 0.0f;
}

__global__ void pool_sum(const float* __restrict__ h2, const int* __restrict__ batch,
                         float* __restrict__ gsum, float* __restrict__ gcnt, int nNodes)
{
    const int idx = blockIdx.x * blockDim.x + threadIdx.x;   // over nNodes*32
    if (idx >= nNodes * 32) return;
    const int n = idx >> 5, d = idx & 31;
    const int g = batch[n];
    atomicAdd(&gsum[g * 32 + d], h2[idx]);
    if (d == 0) atomicAdd(&gcnt[g], 1.0f);
}

// ---------------- fusion head (one workgroup per graph) --------------------
__global__ void fusion_head(const float* __restrict__ gsum, const float* __restrict__ gcnt,
                            const float* __restrict__ obs, const float* __restrict__ noise,
                            const float* __restrict__ nexp,
                            const float* __restrict__ Wo1, const float* __restrict__ bo1,
                            const float* __restrict__ Wo2, const float* __restrict__ bo2,
                            const float* __restrict__ Wn,  const float* __restrict__ bn,
                            const float* __restrict__ Wf1, const float* __restrict__ bf1,
                            const float* __restrict__ Wf2, const float* __restrict__ bf2,
                            const float* __restrict__ Wf3, const float* __restrict__ bf3,
                            float* __restrict__ out)
{
    __shared__ float comb[45];
    __shared__ float h1s[256];
    __shared__ float h2s[32];
    const int g = blockIdx.x, t = threadIdx.x;

    if (t < 32) {                       // circuit embedding: mean pool
        float cnt = gcnt[g];
        cnt = (cnt > 1.0f) ? cnt : 1.0f;
        comb[t] = gsum[g * 32 + t] / cnt;
    } else if (t < 40) {                // obs encoder: 5 -> relu(32) -> 8
        const int j = t - 32;
        float acc = bo2[j];
        for (int k = 0; k < 32; ++k) {
            float o = bo1[k];
            for (int c = 0; c < 5; ++c) o += obs[g * 5 + c] * Wo1[c * 32 + k];
            o = (o > 0.0f) ? o : 0.0f;
            acc += o * Wo2[k * 8 + j];
        }
        comb[t] = acc;
    } else if (t < 44) {                // noise encoder: 1 -> 4
        const int j = t - 40;
        comb[t] = noise[g] * Wn[j] + bn[j];
    } else if (t == 44) {
        comb[44] = nexp[g];
    }
    __syncthreads();

    {   // hf1 = relu(comb @ Wf1 + bf1), 256-wide; thread t computes one output
        float acc = bf1[t];
        for (int k = 0; k < 45; ++k) acc += comb[k] * Wf1[k * 256 + t];
        h1s[t] = (acc > 0.0f) ? acc : 0.0f;
    }
    __syncthreads();
    if (t < 32) {                       // hf2 = relu(hf1 @ Wf2 + bf2)
        float acc = bf2[t];
        for (int k = 0; k < 256; ++k) acc += h1s[k] * Wf2[k * 32 + t];
        h2s[t] = (acc > 0.0f) ? acc : 0.0f;
    }
    __syncthreads();
    if (t == 0) {                       // correction + residual
        float acc = bf3[0];
        for (int k = 0; k < 32; ++k) acc += h2s[k] * Wf3[k];
        out[g] = nexp[g] + acc;
    }
}

// ---------------------------------------------------------------------------
extern "C" void kernel_launch(void* const* d_in, const int* in_sizes, int n_in,
                              void* d_out, int out_size, void* d_ws, size_t ws_size,
                              hipStream_t stream)
{
    const float* x     = (const float*)d_in[0];
    const int*   ei    = (const int*)  d_in[1];   // [2, 800000]
    const int*   batch = (const int*)  d_in[2];
    const float* obs   = (const float*)d_in[3];
    const float* noise = (const float*)d_in[4];
    const float* nexp  = (const float*)d_in[5];
    const float* Wl1   = (const float*)d_in[6];
    const float* bl1   = (const float*)d_in[7];
    const float* Wr1   = (const float*)d_in[8];
    const float* br1   = (const float*)d_in[9];
    const float* att1  = (const float*)d_in[10];
    const float* bias1 = (const float*)d_in[11];
    const float* Wl2   = (const float*)d_in[12];
    const float* bl2   = (const float*)d_in[13];
    const float* Wr2   = (const float*)d_in[14];
    const float* br2   = (const float*)d_in[15];
    const float* att2  = (const float*)d_in[16];
    const float* bias2 = (const float*)d_in[17];
    const float* Wo1   = (const float*)d_in[18];
    const float* bo1   = (const float*)d_in[19];
    const float* Wo2   = (const float*)d_in[20];
    const float* bo2   = (const float*)d_in[21];
    const float* Wn    = (const float*)d_in[22];
    const float* bn    = (const float*)d_in[23];
    const float* Wf1   = (const float*)d_in[24];
    const float* bf1   = (const float*)d_in[25];
    const float* Wf2   = (const float*)d_in[26];
    const float* bf2   = (const float*)d_in[27];
    const float* Wf3   = (const float*)d_in[28];
    const float* bf3   = (const float*)d_in[29];
    float* out = (float*)d_out;

    const int N = N_NODES, E = N_EDGES, ET = N_EDGES + N_NODES, G = N_GRAPHS;

    // ---- workspace carve-up (floats); peak ~81 MB ----
    float* ws   = (float*)d_ws;
    float* xl1  = ws;                         // N*128
    float* xr1  = xl1  + (size_t)N * 128;     // N*128
    float* h1   = xr1  + (size_t)N * 128;     // N*128
    float* esc  = h1   + (size_t)N * 128;     // ET
    float* nmax = esc  + ET;                  // N
    float* nden = nmax + N;                   // N
    float* gsum = nden + N;                   // G*32
    float* gcnt = gsum + (size_t)G * 32;      // G
    // layer-2 buffers alias the (now free) xl1 region
    float* xl2  = xl1;                        // N*32
    float* xr2  = xl1 + (size_t)N * 32;       // N*32
    float* h2   = xl1 + (size_t)2 * N * 32;   // N*32

    // ================= Layer 1: 17 -> 128 =================
    init_layer<<<(N * 128 + 255) / 256, 256, 0, stream>>>(nmax, nden, h1, N, 128);
    gemm_wmma_lr<17, 128><<<dim3(N / 16, 128 / 16), 32, 0, stream>>>(
        x, Wl1, bl1, Wr1, br1, xl1, xr1);
    edge_score<128><<<(ET + 7) / 8, 256, 0, stream>>>(xl1, xr1, att1, ei, esc, nmax, E, N);
    edge_exp<<<(ET + 255) / 256, 256, 0, stream>>>(ei, nmax, esc, nden, E, N);
    edge_aggregate<128><<<(ET + 7) / 8, 256, 0, stream>>>(xl1, ei, esc, nden, h1, E, N);
    bias_elu4<<<(N * 128 / 4 + 255) / 256, 256, 0, stream>>>((float4*)h1, bias1, N * 128 / 4, 128);

    // ================= Layer 2: 128 -> 32 =================
    init_layer<<<(N * 32 + 255) / 256, 256, 0, stream>>>(nmax, nden, h2, N, 32);
    gemm_wmma_lr<128, 32><<<dim3(N / 16, 32 / 16), 32, 0, stream>>>(
        h1, Wl2, bl2, Wr2, br2, xl2, xr2);
    edge_score<32><<<(ET + 7) / 8, 256, 0, stream>>>(xl2, xr2, att2, ei, esc, nmax, E, N);
    edge_exp<<<(ET + 255) / 256, 256, 0, stream>>>(ei, nmax, esc, nden, E, N);
    edge_aggregate<32><<<(ET + 7) / 8, 256, 0, stream>>>(xl2, ei, esc, nden, h2, E, N);
    bias_elu4<<<(N * 32 / 4 + 255) / 256, 256, 0, stream>>>((float4*)h2, bias2, N * 32 / 4, 32);

    // ================= Pool + heads =================
    init_pool<<<(G * 32 + 255) / 256, 256, 0, stream>>>(gsum, gcnt, G);
    pool_sum<<<(N * 32 + 255) / 256, 256, 0, stream>>>(h2, batch, gsum, gcnt, N);
    fusion_head<<<G, 256, 0, stream>>>(gsum, gcnt, obs, noise, nexp,
                                       Wo1, bo1, Wo2, bo2, Wn, bn,
                                       Wf1, bf1, Wf2, bf2, Wf3, bf3, out);
}